// LstmModel_31937376813559
// MI455X (gfx1250) — compile-verified
//
#include <hip/hip_runtime.h>
#include <math.h>

#define HDIM 512
#define GDIM 2048            // 4*H
#define TLEN 4096
#define SCAN_NWG 64
#define WPAD 528             // 512 + 16 -> conflict-free LDS banking

typedef float v2f __attribute__((ext_vector_type(2)));
typedef float v8f __attribute__((ext_vector_type(8)));

__device__ __forceinline__ float sigmoidf_(float x) {
    return 1.0f / (1.0f + expf(-x));
}

// ---------------------------------------------------------------------------
// XP[T][2048] = X[T][K] @ Wih^T + (bih + bhh), fp32 WMMA 16x16x4.
// 64x64 C-tile per workgroup, K staged through LDS in chunks of 32.
// 8 waves: wave w -> N-subtile (w&3), M-subtiles {2*(w>>2), 2*(w>>2)+1}.
// ---------------------------------------------------------------------------
__global__ __launch_bounds__(256) void xproj_wmma_kernel(
    const float* __restrict__ X, const float* __restrict__ Wih,
    const float* __restrict__ bih, const float* __restrict__ bhh,
    float* __restrict__ XP, int K)
{
    __shared__ float Xs[64][33];
    __shared__ float Ws[64][33];

    const int tid  = threadIdx.x;
    const int wave = tid >> 5;
    const int lane = tid & 31;
    const int ln   = lane & 15;
    const int hi   = lane >> 4;          // half-wave select (K pairs / M+8)
    const int m0   = blockIdx.x * 64;
    const int n0   = blockIdx.y * 64;
    const int sn   = wave & 3;
    const int sm0  = (wave >> 2) * 2;

    v8f acc0 = {};
    v8f acc1 = {};

    for (int k0 = 0; k0 < K; k0 += 32) {
        for (int i = tid; i < 64 * 32; i += 256) {
            int r = i >> 5, c = i & 31;
            Xs[r][c] = X  [(size_t)(m0 + r) * K + (k0 + c)];
            Ws[r][c] = Wih[(size_t)(n0 + r) * K + (k0 + c)];
        }
        if (k0 + 32 < K) {   // hint next chunk -> global_prefetch_b8
            __builtin_prefetch(&X  [(size_t)(m0 + (tid >> 5)) * K + k0 + 32 + (tid & 31)], 0, 1);
            __builtin_prefetch(&Wih[(size_t)(n0 + (tid >> 5)) * K + k0 + 32 + (tid & 31)], 0, 1);
        }
        __syncthreads();

        #pragma unroll
        for (int kk = 0; kk < 32; kk += 4) {
            const int kc = kk + 2 * hi;   // lanes 0-15: K={kk,kk+1}; 16-31: K={kk+2,kk+3}
            v2f a0, a1, b;
            a0.x = Xs[sm0 * 16 + ln][kc];       a0.y = Xs[sm0 * 16 + ln][kc + 1];
            a1.x = Xs[(sm0 + 1) * 16 + ln][kc]; a1.y = Xs[(sm0 + 1) * 16 + ln][kc + 1];
            b.x  = Ws[sn * 16 + ln][kc];        b.y  = Ws[sn * 16 + ln][kc + 1];
            acc0 = __builtin_amdgcn_wmma_f32_16x16x4_f32(false, a0, false, b,
                                                         (short)0, acc0, false, false);
            acc1 = __builtin_amdgcn_wmma_f32_16x16x4_f32(false, a1, false, b,
                                                         (short)0, acc1, false, false);
        }
        __syncthreads();
    }

    // C/D layout: VGPR v, lanes 0-15 -> M=v,N=lane; lanes 16-31 -> M=v+8,N=lane-16
    const int n = n0 + sn * 16 + ln;
    const float bias = bih[n] + bhh[n];
    #pragma unroll
    for (int v = 0; v < 8; ++v) {
        int m = m0 + sm0 * 16 + v + 8 * hi;
        XP[(size_t)m * GDIM + n] = acc0[v] + bias;
        XP[(size_t)(m + 16) * GDIM + n] = acc1[v] + bias;
    }
}

// ---------------------------------------------------------------------------
// Sequential LSTM scan: 64 persistent WGs, each owns 8 h-indices and the
// 32 matching gate rows of Whh (LDS-resident, fp32, pitch 528 = conflict-free).
// Per timestep: 32x512 matvec from LDS, gate nonlinearity, write h slice,
// device-scope monotonic barrier (atomic counter + s_sleep spin).
// ---------------------------------------------------------------------------
__global__ __launch_bounds__(256) void lstm_scan_kernel(
    const float* __restrict__ XP, const float* __restrict__ Whh,
    float* __restrict__ Hout, unsigned* bar)
{
    extern __shared__ float smem[];
    float* Wl    = smem;                 // 32 * WPAD
    float* hbuf  = Wl + 32 * WPAD;       // 512
    float* gates = hbuf + HDIM;          // 32
    float* cbuf  = gates + 32;           // 8

    const int tid = threadIdx.x;
    const int hb  = blockIdx.x * 8;      // owned h-index base

    // preload this WG's 32 gate rows of Whh into LDS (coalesced)
    for (int idx = tid; idx < 32 * HDIM; idx += 256) {
        int lr  = idx >> 9;              // local row 0..31: gate = lr>>3, j = lr&7
        int col = idx & 511;
        int grow = (lr >> 3) * HDIM + hb + (lr & 7);
        Wl[lr * WPAD + col] = Whh[(size_t)grow * HDIM + col];
    }
    if (tid < 8) cbuf[tid] = 0.0f;
    __syncthreads();

    const int r     = tid >> 3;          // local gate row per thread group
    const int chunk = tid & 7;           // 8 threads per row
    const int grow  = (r >> 3) * HDIM + hb + (r & 7);

    for (int t = 0; t < TLEN; ++t) {
        // fetch h_{t-1} (full 512) into LDS
        if (t == 0) {
            for (int j = tid; j < HDIM; j += 256) hbuf[j] = 0.0f;
        } else {
            const float* hp = Hout + (size_t)(t - 1) * HDIM;
            for (int j = tid; j < HDIM; j += 256) hbuf[j] = hp[j];
        }
        __syncthreads();

        // 512-dot split over 8 threads; strided elems -> bank-conflict-free
        float acc = 0.0f;
        const float* wr = Wl + r * WPAD;
        #pragma unroll 8
        for (int k = 0; k < 64; ++k) {
            int e = chunk + (k << 3);
            acc += wr[e] * hbuf[e];
        }
        acc += __shfl_xor(acc, 4, 32);
        acc += __shfl_xor(acc, 2, 32);
        acc += __shfl_xor(acc, 1, 32);
        if (chunk == 0)
            gates[r] = XP[(size_t)t * GDIM + grow] + acc;
        __syncthreads();

        if (tid < 8) {  // gate order i, f, g, o (PyTorch)
            float ig = sigmoidf_(gates[tid]);
            float fg = sigmoidf_(gates[8 + tid]);
            float gg = tanhf(gates[16 + tid]);
            float og = sigmoidf_(gates[24 + tid]);
            float c  = fg * cbuf[tid] + ig * gg;
            cbuf[tid] = c;
            Hout[(size_t)t * HDIM + hb + tid] = og * tanhf(c);
        }
        __syncthreads();

        // device-wide step barrier (monotonic counter; reset by reset_bar_kernel)
        if (tid == 0) {
            __threadfence();
            atomicAdd(bar, 1u);
            const unsigned target = (unsigned)SCAN_NWG * (unsigned)(t + 1);
            while (__hip_atomic_load(bar, __ATOMIC_RELAXED, __HIP_MEMORY_SCOPE_AGENT) < target) {
                __builtin_amdgcn_s_sleep(2);
            }
        }
        __syncthreads();
        __threadfence();
    }
}

__global__ void reset_bar_kernel(unsigned* bar) { *bar = 0u; }

// ---------------------------------------------------------------------------
// MLP head: one wave per row. fc1(512->64)+silu, fc2(64->32)+silu, fc3(32->16)
// ---------------------------------------------------------------------------
__global__ __launch_bounds__(256) void head_kernel(
    const float* __restrict__ Hseq,
    const float* __restrict__ fc1w, const float* __restrict__ fc1b,
    const float* __restrict__ fc2w, const float* __restrict__ fc2b,
    const float* __restrict__ fc3w, const float* __restrict__ fc3b,
    float* __restrict__ out)
{
    __shared__ float o1[8][64];
    __shared__ float o2[8][32];
    const int tid  = threadIdx.x;
    const int wave = tid >> 5;
    const int lane = tid & 31;
    const int row  = blockIdx.x * 8 + wave;
    const float* h = Hseq + (size_t)row * HDIM;

    float a0 = 0.f, a1 = 0.f;
    for (int k = 0; k < HDIM; ++k) {
        float hv = h[k];
        a0 += hv * fc1w[(size_t)lane * HDIM + k];
        a1 += hv * fc1w[(size_t)(lane + 32) * HDIM + k];
    }
    a0 += fc1b[lane];      a1 += fc1b[lane + 32];
    o1[wave][lane]      = a0 * sigmoidf_(a0);
    o1[wave][lane + 32] = a1 * sigmoidf_(a1);
    __syncthreads();

    float b0 = 0.f;
    for (int j = 0; j < 64; ++j) b0 += o1[wave][j] * fc2w[lane * 64 + j];
    b0 += fc2b[lane];
    o2[wave][lane] = b0 * sigmoidf_(b0);
    __syncthreads();

    if (lane < 16) {
        float c0 = 0.f;
        for (int j = 0; j < 32; ++j) c0 += o2[wave][j] * fc3w[lane * 32 + j];
        out[(size_t)row * 16 + lane] = c0 + fc3b[lane];
    }
}

// ---------------------------------------------------------------------------
extern "C" void kernel_launch(void* const* d_in, const int* in_sizes, int n_in,
                              void* d_out, int out_size, void* d_ws, size_t ws_size,
                              hipStream_t stream)
{
    (void)in_sizes; (void)n_in; (void)out_size; (void)ws_size;

    const float* x     = (const float*)d_in[0];
    const float* eWih0 = (const float*)d_in[1];
    const float* eWhh0 = (const float*)d_in[2];
    const float* ebih0 = (const float*)d_in[3];
    const float* ebhh0 = (const float*)d_in[4];
    const float* eWih  = (const float*)d_in[5];
    const float* eWhh  = (const float*)d_in[6];
    const float* ebih  = (const float*)d_in[7];
    const float* ebhh  = (const float*)d_in[8];
    const float* dWih  = (const float*)d_in[9];
    const float* dWhh  = (const float*)d_in[10];
    const float* dbih  = (const float*)d_in[11];
    const float* dbhh  = (const float*)d_in[12];
    const float* fc1w  = (const float*)d_in[13];
    const float* fc1b  = (const float*)d_in[14];
    const float* fc2w  = (const float*)d_in[15];
    const float* fc2b  = (const float*)d_in[16];
    const float* fc3w  = (const float*)d_in[17];
    const float* fc3b  = (const float*)d_in[18];

    char*  ws   = (char*)d_ws;
    float* XP   = (float*)ws;                                   // T*2048 f32 (32 MB)
    float* seqA = (float*)(ws + (size_t)TLEN * GDIM * 4);       // T*512 f32 (8 MB)
    float* seqB = seqA + (size_t)TLEN * HDIM;                   // T*512 f32 (8 MB)
    unsigned* bar = (unsigned*)(seqB + (size_t)TLEN * HDIM);

    const dim3 ggrid(TLEN / 64, GDIM / 64);
    const size_t scan_lds = (size_t)(32 * WPAD + HDIM + 32 + 8) * sizeof(float);

    const float* in = x;
    int K = 128;
    float* outbuf = seqA;

    for (int layer = 0; layer < 16; ++layer) {
        const float *Wih, *Whh, *bih, *bhh;
        if (layer == 0) {
            Wih = eWih0; Whh = eWhh0; bih = ebih0; bhh = ebhh0;
        } else if (layer < 8) {
            size_t o = (size_t)(layer - 1);
            Wih = eWih + o * GDIM * HDIM; Whh = eWhh + o * GDIM * HDIM;
            bih = ebih + o * GDIM;        bhh = ebhh + o * GDIM;
        } else {
            size_t o = (size_t)(layer - 8);
            Wih = dWih + o * GDIM * HDIM; Whh = dWhh + o * GDIM * HDIM;
            bih = dbih + o * GDIM;        bhh = dbhh + o * GDIM;
        }
        reset_bar_kernel<<<1, 1, 0, stream>>>(bar);
        xproj_wmma_kernel<<<ggrid, 256, 0, stream>>>(in, Wih, bih, bhh, XP, K);
        lstm_scan_kernel<<<SCAN_NWG, 256, scan_lds, stream>>>(XP, Whh, outbuf, bar);
        in = outbuf;
        K = HDIM;
        outbuf = (outbuf == seqA) ? seqB : seqA;
    }

    head_kernel<<<TLEN / 8, 256, 0, stream>>>(in, fc1w, fc1b, fc2w, fc2b,
                                              fc3w, fc3b, (float*)d_out);
}